// GCN_42417097015690
// MI455X (gfx1250) — compile-verified
//
#include <hip/hip_runtime.h>
#include <hip/hip_bf16.h>

typedef __attribute__((ext_vector_type(2))) float v2f;
typedef __attribute__((ext_vector_type(8))) float v8f;

#define N_NODES 100000
#define N_EDGES 3200000
#define FEATS   256
#define NCLS    40
#define NPAD    48   // 40 classes padded to 3 full 16-col WMMA tiles

// ---------------------------------------------------------------------------
// Zero-fill (bufA must be zeroed before every scatter-accumulate pass).
// ---------------------------------------------------------------------------
__global__ void zero_f32(float* __restrict__ p, long long n) {
  long long i = (long long)blockIdx.x * blockDim.x + threadIdx.x;
  long long stride = (long long)gridDim.x * blockDim.x;
  for (; i < n; i += stride) p[i] = 0.0f;
}

// ---------------------------------------------------------------------------
// Pad W3 [256,40] -> Wp [256,48], zero-filling cols 40..47.
// ---------------------------------------------------------------------------
__global__ void pad_w3(const float* __restrict__ W3, float* __restrict__ Wp) {
  int i = blockIdx.x * blockDim.x + threadIdx.x;
  if (i >= FEATS * NPAD) return;
  int r = i / NPAD;
  int c = i - r * NPAD;
  Wp[i] = (c < NCLS) ? W3[r * NCLS + c] : 0.0f;
}

// ---------------------------------------------------------------------------
// SpMM: out[dst] += val * H[src], feature dim = 256.
// Block (64,4): threadIdx.x = feature-quad, threadIdx.y = edge in block.
// 1KB coalesced gather per edge; H_in/H_out are ~L2-resident (192MB L2).
// Scatter via global_atomic_add_f32.
// ---------------------------------------------------------------------------
__global__ void spmm256(const int* __restrict__ ei,     // [2, E]
                        const float* __restrict__ vals, // [E]
                        const float* __restrict__ H,    // [N, 256]
                        float* __restrict__ out,        // [N, 256] (pre-zeroed)
                        int nedges) {
  int e = blockIdx.x * blockDim.y + threadIdx.y;
  if (e >= nedges) return;
  int s = ei[e];
  int d = ei[nedges + e];
  float v = vals[e];
  int f = threadIdx.x << 2;
  const float4 h = *(const float4*)(H + (long long)s * FEATS + f);
  float* op = out + (long long)d * FEATS + f;
  atomicAdd(op + 0, v * h.x);
  atomicAdd(op + 1, v * h.y);
  atomicAdd(op + 2, v * h.z);
  atomicAdd(op + 3, v * h.w);
}

// ---------------------------------------------------------------------------
// C[M,N] = A[M,K] * W[K,N] (+ optional ReLU) with V_WMMA_F32_16X16X4_F32.
// Compile-time N (no ragged columns -> no divergence, immediate-offset
// addressing). Each wave computes NT adjacent 16x16 tiles (16 x 16*NT strip):
// one b64 A-load is shared by NT independent WMMAs per k-step, and the NT
// accumulators break the WMMA->WMMA accumulation dependency chain.
// Requirements: M % 16 == 0, K % 4 == 0, N % (16*NT) == 0.
//
// Layouts (ISA 7.12.2): A 16x4: lane<16 -> M=lane,K0/K1; lane>=16 -> K2/K3.
// B 4x16: VGPR0 rows {K0,K2}, VGPR1 rows {K1,K3}, N across lanes.
// C/D: VGPR r -> M = r + 8*(lane>=16), N = lane&15.
// Only wave-uniform branching: EXEC is all-ones at every WMMA.
// ---------------------------------------------------------------------------
template <int N, int NT, int RELU>
__global__ void gemm_wmma(const float* __restrict__ A,
                          const float* __restrict__ W,
                          float* __restrict__ C, int M, int K) {
  constexpr int NG = N / (16 * NT);        // strip groups along N
  const int waves_per_block = blockDim.x >> 5;
  const int wave = threadIdx.x >> 5;
  const int lane = threadIdx.x & 31;

  const int g = blockIdx.x * waves_per_block + wave;
  const int ntm = M >> 4;
  if (g >= ntm * NG) return;               // wave-uniform: EXEC untouched

  const int mt = g / NG;
  const int ng = g - mt * NG;
  const int m0 = mt << 4;
  const int n0 = ng * (16 * NT);

  const int l15   = lane & 15;
  const int khalf = (lane >> 4) << 1;      // 0 (lanes 0-15) or 2 (lanes 16-31)

  const float* Arow = A + (size_t)(m0 + l15) * K + khalf;
  const float* Wc   = W + n0 + l15;        // tile t adds +16*t (imm offset)

  v8f c[NT];
#pragma unroll
  for (int t = 0; t < NT; ++t) c[t] = (v8f){};

#pragma unroll 2
  for (int k = 0; k < K; k += 4) {
    const v2f a = *(const v2f*)(Arow + k);           // A[m][k+khalf..+1]
    const float* wk0 = Wc + (size_t)(k + khalf) * N; // row k+khalf
    const float* wk1 = wk0 + N;                      // row k+khalf+1
#pragma unroll
    for (int t = 0; t < NT; ++t) {
      v2f b;
      b.x = wk0[16 * t];
      b.y = wk1[16 * t];
      c[t] = __builtin_amdgcn_wmma_f32_16x16x4_f32(
          false, a, false, b, (short)0, c[t], false, false);
    }
  }

  const int rbase = m0 + ((lane >> 4) << 3);
#pragma unroll
  for (int t = 0; t < NT; ++t) {
    const int col = n0 + 16 * t + l15;
    float* Cc = C + (size_t)rbase * N + col;
#pragma unroll
    for (int r = 0; r < 8; ++r) {
      float v = c[t][r];
      if (RELU) v = v > 0.0f ? v : 0.0f;
      Cc[(size_t)r * N] = v;
    }
  }
}

// ---------------------------------------------------------------------------
// Row-wise log_softmax over 40 classes; input rows padded to stride 48.
// ---------------------------------------------------------------------------
__global__ void logsoftmax40(const float* __restrict__ logits,  // [rows, 48]
                             float* __restrict__ out,           // [rows, 40]
                             int rows) {
  int r = blockIdx.x * blockDim.x + threadIdx.x;
  if (r >= rows) return;
  const float* x = logits + (long long)r * NPAD;
  float m = -INFINITY;
#pragma unroll
  for (int j = 0; j < NCLS; ++j) m = fmaxf(m, x[j]);
  float s = 0.0f;
#pragma unroll
  for (int j = 0; j < NCLS; ++j) s += expf(x[j] - m);
  const float ls = logf(s);
  float* o = out + (long long)r * NCLS;
#pragma unroll
  for (int j = 0; j < NCLS; ++j) o[j] = x[j] - m - ls;
}

// ---------------------------------------------------------------------------
extern "C" void kernel_launch(void* const* d_in, const int* in_sizes, int n_in,
                              void* d_out, int out_size, void* d_ws, size_t ws_size,
                              hipStream_t stream) {
  const float* x  = (const float*)d_in[0];   // [N_NODES, 256]
  const int*   ei = (const int*)d_in[1];     // [2, N_EDGES]
  const float* ev = (const float*)d_in[2];   // [N_EDGES]
  const float* W1 = (const float*)d_in[3];   // [256, 256]
  const float* W2 = (const float*)d_in[4];   // [256, 256]
  const float* W3 = (const float*)d_in[5];   // [256, 40]
  float* out = (float*)d_out;                // [N_NODES, 40]

  float* bufA   = (float*)d_ws;                          // [N_NODES,256]
  float* bufB   = bufA + (size_t)N_NODES * FEATS;        // [N_NODES,256]
  float* logits = bufB + (size_t)N_NODES * FEATS;        // [N_NODES,48]
  float* Wpad   = logits + (size_t)N_NODES * NPAD;       // [256,48]

  const long long nA = (long long)N_NODES * FEATS;
  const dim3 sb(64, 4);
  const int  sgrid = (N_EDGES + 3) / 4;

  const int GB = 256;                                    // 8 waves / block
  // N=256, NT=4: strips = (M/16) * (256/64) = 6250*4
  const int strips256 = (N_NODES / 16) * (FEATS / 64);
  const int ggrid256  = (strips256 + GB / 32 - 1) / (GB / 32);
  // N=48, NT=3: strips = M/16
  const int strips48 = N_NODES / 16;
  const int ggrid48  = (strips48 + GB / 32 - 1) / (GB / 32);

  pad_w3<<<(FEATS * NPAD + 255) / 256, 256, 0, stream>>>(W3, Wpad);

  // Layer 1: relu( (A @ X) @ W1 )
  zero_f32<<<2048, 256, 0, stream>>>(bufA, nA);
  spmm256<<<sgrid, sb, 0, stream>>>(ei, ev, x, bufA, N_EDGES);
  gemm_wmma<FEATS, 4, 1><<<ggrid256, GB, 0, stream>>>(bufA, W1, bufB, N_NODES, FEATS);

  // Layer 2: relu( (A @ H) @ W2 )
  zero_f32<<<2048, 256, 0, stream>>>(bufA, nA);
  spmm256<<<sgrid, sb, 0, stream>>>(ei, ev, bufB, bufA, N_EDGES);
  gemm_wmma<FEATS, 4, 1><<<ggrid256, GB, 0, stream>>>(bufA, W2, bufB, N_NODES, FEATS);

  // Layer 3: (A @ H) @ W3pad -> logits[.,48], then log_softmax on first 40
  zero_f32<<<2048, 256, 0, stream>>>(bufA, nA);
  spmm256<<<sgrid, sb, 0, stream>>>(ei, ev, bufB, bufA, N_EDGES);
  gemm_wmma<NPAD, 3, 0><<<ggrid48, GB, 0, stream>>>(bufA, Wpad, logits, N_NODES, FEATS);
  logsoftmax40<<<(N_NODES + 255) / 256, 256, 0, stream>>>(logits, out, N_NODES);
}